// Loss_20212116095273
// MI455X (gfx1250) — compile-verified
//
#include <hip/hip_runtime.h>
#include <math.h>

// ---------------------------------------------------------------------------
// Problem constants (match reference)
// ---------------------------------------------------------------------------
#define PTS   200000        // P
#define TT    16            // T
#define FDIR  15            // F_DIR
#define BLK   256           // 8 waves of 32 on CDNA5
#define GRID1 2048          // partial-sum blocks for direction loss (mult of 64)
#define GRID3 4096          // blocks for output kernel

typedef float v2f __attribute__((ext_vector_type(2)));
typedef float v8f __attribute__((ext_vector_type(8)));

// conv_bbox: (a,b,c,d) -> (a-0.5c, b-0.5d, a, b)
__device__ __forceinline__ float4 conv_bbox(float4 b) {
    return make_float4(b.x - 0.5f * b.z, b.y - 0.5f * b.w, b.x, b.y);
}

// corner_pts: 5 (x,y) pairs from (x0,y0,x1,y1)
__device__ __forceinline__ void corner_pts(float4 b, float xs[5], float ys[5]) {
    xs[0] = 0.5f * (b.x + b.z); ys[0] = 0.5f * (b.y + b.w);
    xs[1] = b.x;                ys[1] = b.y;
    xs[2] = b.x;                ys[2] = b.w;
    xs[3] = b.z;                ys[3] = b.y;
    xs[4] = b.z;                ys[4] = b.w;
}

// ---------------------------------------------------------------------------
// Kernel 1: per-block partial sums of the direction-loss arccos terms.
// One item = one (f, p) pair, f in [0,15), p in [0,P).
// outputs viewed as float4[T*P]; targets viewed as float4[T*P*2]
// (targets[t,p,0:4] = tgt4[(t*P+p)*2], targets[t,p,4:8] = tgt4[(t*P+p)*2+1]).
// ---------------------------------------------------------------------------
__global__ void dir_partial_kernel(const float4* __restrict__ out4,
                                   const float4* __restrict__ tgt4,
                                   float* __restrict__ partials) {
    const int N      = FDIR * PTS;
    const int stride = gridDim.x * blockDim.x;
    float acc = 0.0f;

    for (int idx = blockIdx.x * blockDim.x + threadIdx.x; idx < N; idx += stride) {
        // Prefetch one grid-stride ahead (global_prefetch_b8; speculative, OOB
        // addresses are silently dropped by the translation path).
        {
            int nidx = idx + stride;
            const int f2 = nidx / PTS;
            __builtin_prefetch(&out4[nidx], 0, 0);
            __builtin_prefetch(&tgt4[(size_t)nidx * 2], 0, 0);
            __builtin_prefetch(&tgt4[((size_t)(f2 + 1) * PTS + (nidx - f2 * PTS)) * 2], 0, 0);
        }

        const int f = idx / PTS;
        const int p = idx - f * PTS;

        float4 L  = tgt4[((size_t)f * PTS + p) * 2];           // targets[f,p,0:4]
        float4 Tr = tgt4[((size_t)(f + 1) * PTS + p) * 2];     // targets[f+1,p,0:4]
        float4 O  = out4[(size_t)f * PTS + p];                 // outputs[f,p,:]

        float4 pred = make_float4(L.x + O.x, L.y + O.y, L.z + O.z, L.w + O.w);

        float4 lc1  = conv_bbox(L);
        float4 leff = (f == 0) ? lc1 : conv_bbox(lc1);
        float4 pc   = conv_bbox(pred);
        float4 tc   = conv_bbox(Tr);

        float lx[5], ly[5], px[5], py[5], tx[5], ty[5];
        corner_pts(leff, lx, ly);
        corner_pts(pc,   px, py);
        corner_pts(tc,   tx, ty);

#pragma unroll
        for (int k = 0; k < 5; ++k) {
            float pdx = px[k] - lx[k], pdy = py[k] - ly[k];
            float tdx = tx[k] - lx[k], tdy = ty[k] - ly[k];
            float pn = sqrtf(pdx * pdx + pdy * pdy) + 1e-6f;
            float tn = sqrtf(tdx * tdx + tdy * tdy) + 1e-6f;
            float cs = (pdy * tdy + pdx * tdx) / (pn * tn);
            cs = fminf(1.0f, fmaxf(-1.0f, cs));
            acc += acosf(cs);
        }
    }

    // wave32 shuffle reduction
#pragma unroll
    for (int m = 16; m >= 1; m >>= 1)
        acc += __shfl_xor(acc, m, 32);

    __shared__ float ws[BLK / 32];
    const int lane = threadIdx.x & 31;
    const int wid  = threadIdx.x >> 5;
    if (lane == 0) ws[wid] = acc;
    __syncthreads();

    if (threadIdx.x == 0) {
        float s = 0.0f;
#pragma unroll
        for (int w = 0; w < BLK / 32; ++w) s += ws[w];
        partials[blockIdx.x] = s;
    }
}

// ---------------------------------------------------------------------------
// Kernel 2: reduce GRID1 partials to the final scalar 0.01 * loss_dir using
// V_WMMA_F32_16X16X4_F32: C += ones(16x4) x B(4x16). Each WMMA consumes 64
// partials (one global_load_b64 per lane, no guards: GRID1 is a compile-time
// multiple of 64). Every element of C is a column sum, so summing C's VGPR0
// across all 32 lanes yields 2x the grand total. One wave, EXEC all ones.
// ---------------------------------------------------------------------------
__global__ void dir_finalize_kernel(const float* __restrict__ partials,
                                    float* __restrict__ scalar_out) {
    const v2f* __restrict__ p2 = (const v2f*)partials;   // GRID1/2 v2f elements
    const int lane = threadIdx.x;                        // 0..31
    v8f c = {};
    v2f ones; ones.x = 1.0f; ones.y = 1.0f;              // A = 16x4 all-ones

#pragma unroll 4
    for (int base = 0; base < GRID1 / 2; base += 32) {
        v2f b = p2[base + lane];                         // 64 partials per wave
        c = __builtin_amdgcn_wmma_f32_16x16x4_f32(
                /*neg_a=*/false, ones, /*neg_b=*/false, b,
                /*c_mod=*/(short)0, c, /*reuse_a=*/false, /*reuse_b=*/false);
    }

    float t = c[0];                 // VGPR0: lanes 0..15 = C[0][n], 16..31 = C[8][n]
#pragma unroll
    for (int m = 16; m >= 1; m >>= 1)
        t += __shfl_xor(t, m, 32);  // = 2 * grand total

    if (lane == 0) {
        // 0.01 * (0.2 * total / (P * F_DIR)), with the x2 undone
        scalar_out[0] = 0.5f * t * (0.01f * 0.2f / ((float)PTS * (float)FDIR));
    }
}

// ---------------------------------------------------------------------------
// Kernel 3: out[t,p] = 0.99 * (sum_c smoothL1(outputs-delta)) / P  + 0.01*loss_dir
// ---------------------------------------------------------------------------
__device__ __forceinline__ float sl1(float d) {
    float ad = fabsf(d);
    return (ad < 1.0f) ? (0.5f * d * d) : (ad - 0.5f);
}

__global__ void loss_out_kernel(const float4* __restrict__ out4,
                                const float4* __restrict__ tgt4,
                                const float* __restrict__ dir_scalar,
                                float* __restrict__ out) {
    const int N      = TT * PTS;
    const int stride = gridDim.x * blockDim.x;
    const float dscaled = dir_scalar[0];          // already 0.01 * loss_dir
    const float invP = 1.0f / (float)PTS;

    for (int idx = blockIdx.x * blockDim.x + threadIdx.x; idx < N; idx += stride) {
        {
            int nidx = idx + stride;
            __builtin_prefetch(&out4[nidx], 0, 0);
            __builtin_prefetch(&tgt4[(size_t)nidx * 2 + 1], 0, 0);
        }
        float4 O = out4[idx];
        float4 D = tgt4[(size_t)idx * 2 + 1];     // targets[t,p,4:8]
        float s = sl1(O.x - D.x) + sl1(O.y - D.y) + sl1(O.z - D.z) + sl1(O.w - D.w);
        out[idx] = 0.99f * (s * invP) + dscaled;
    }
}

// ---------------------------------------------------------------------------
// Launch
// ---------------------------------------------------------------------------
extern "C" void kernel_launch(void* const* d_in, const int* in_sizes, int n_in,
                              void* d_out, int out_size, void* d_ws, size_t ws_size,
                              hipStream_t stream) {
    const float4* out4 = (const float4*)d_in[0];   // outputs (16,200000,4) f32
    const float4* tgt4 = (const float4*)d_in[1];   // targets (16,200000,8) f32
    float* out = (float*)d_out;                    // (16,200000) f32

    float* partials  = (float*)d_ws;               // GRID1 floats
    float* dirscalar = partials + GRID1;           // 1 float

    dir_partial_kernel<<<GRID1, BLK, 0, stream>>>(out4, tgt4, partials);
    dir_finalize_kernel<<<1, 32, 0, stream>>>(partials, dirscalar);
    loss_out_kernel<<<GRID3, BLK, 0, stream>>>(out4, tgt4, dirscalar, out);
}